// GraphAttentionNetwork_47227460387058
// MI455X (gfx1250) — compile-verified
//
#include <hip/hip_runtime.h>
#include <math.h>

typedef __attribute__((ext_vector_type(2))) float v2f;
typedef __attribute__((ext_vector_type(8))) float v8f;

#define NEG_SLOPE 0.2f
#define BN_EPS 1e-5f

__device__ __forceinline__ float redu16(float v) {
    v += __shfl_xor(v, 1, 32);
    v += __shfl_xor(v, 2, 32);
    v += __shfl_xor(v, 4, 32);
    v += __shfl_xor(v, 8, 32);
    return v;
}

__device__ __forceinline__ void atomicMaxF(float* addr, float v) {
    // ordered-float trick: works because float order == int order for
    // non-negative patterns, and reversed-unsigned order for negative ones.
    if (v >= 0.0f) atomicMax((int*)addr, __float_as_int(v));
    else           atomicMin((unsigned int*)addr, __float_as_uint(v));
}

// ---------------------------------------------------------------------------
// GAT layer 1 GEMM: h1[N,64] = x[N,32] @ W[32,64]; per-head attention dots
// s1[N,4], d1[N,4]. One wave per 16-node tile using V_WMMA_F32_16X16X4_F32.
// ---------------------------------------------------------------------------
__global__ void gat1_gemm(const float* __restrict__ x, const float* __restrict__ W,
                          const float* __restrict__ asrc, const float* __restrict__ adst,
                          float* __restrict__ h1, float* __restrict__ s1,
                          float* __restrict__ d1, int n) {
    int wave = threadIdx.x >> 5;
    int lane = threadIdx.x & 31;
    int tile = blockIdx.x * 8 + wave;
    int ntiles = (n + 15) >> 4;
    if (tile >= ntiles) return;              // wave-uniform: EXEC stays all-1 for WMMA
    int nlane = lane & 15;
    int khalf = lane >> 4;
    int arow = tile * 16 + nlane;
    if (arow >= n) arow = n - 1;             // clamp (n is a multiple of 16 anyway)

    v8f c0 = {}, c1 = {}, c2 = {}, c3 = {};
    #pragma unroll
    for (int kk = 0; kk < 8; ++kk) {
        int k = kk * 4 + khalf * 2;
        v2f a = *(const v2f*)(x + arow * 32 + k);
        v2f b0, b1, b2, b3;
        b0.x = W[k * 64 +  0 + nlane]; b0.y = W[(k + 1) * 64 +  0 + nlane];
        b1.x = W[k * 64 + 16 + nlane]; b1.y = W[(k + 1) * 64 + 16 + nlane];
        b2.x = W[k * 64 + 32 + nlane]; b2.y = W[(k + 1) * 64 + 32 + nlane];
        b3.x = W[k * 64 + 48 + nlane]; b3.y = W[(k + 1) * 64 + 48 + nlane];
        c0 = __builtin_amdgcn_wmma_f32_16x16x4_f32(false, a, false, b0, (short)0, c0, false, false);
        c1 = __builtin_amdgcn_wmma_f32_16x16x4_f32(false, a, false, b1, (short)0, c1, false, false);
        c2 = __builtin_amdgcn_wmma_f32_16x16x4_f32(false, a, false, b2, (short)0, c2, false, false);
        c3 = __builtin_amdgcn_wmma_f32_16x16x4_f32(false, a, false, b3, (short)0, c3, false, false);
    }

    float av0 = asrc[ 0 + nlane], av1 = asrc[16 + nlane];
    float av2 = asrc[32 + nlane], av3 = asrc[48 + nlane];
    float qv0 = adst[ 0 + nlane], qv1 = adst[16 + nlane];
    float qv2 = adst[32 + nlane], qv3 = adst[48 + nlane];

    #pragma unroll
    for (int r = 0; r < 8; ++r) {
        int m = tile * 16 + r + 8 * khalf;   // C layout: VGPR r -> row r + 8*(lane>>4)
        bool ok = (m < n);
        if (ok) {
            h1[m * 64 +  0 + nlane] = c0[r];
            h1[m * 64 + 16 + nlane] = c1[r];
            h1[m * 64 + 32 + nlane] = c2[r];
            h1[m * 64 + 48 + nlane] = c3[r];
        }
        float t;
        t = redu16(c0[r] * av0); if (ok && nlane == 0) s1[m * 4 + 0] = t;
        t = redu16(c1[r] * av1); if (ok && nlane == 0) s1[m * 4 + 1] = t;
        t = redu16(c2[r] * av2); if (ok && nlane == 0) s1[m * 4 + 2] = t;
        t = redu16(c3[r] * av3); if (ok && nlane == 0) s1[m * 4 + 3] = t;
        t = redu16(c0[r] * qv0); if (ok && nlane == 0) d1[m * 4 + 0] = t;
        t = redu16(c1[r] * qv1); if (ok && nlane == 0) d1[m * 4 + 1] = t;
        t = redu16(c2[r] * qv2); if (ok && nlane == 0) d1[m * 4 + 2] = t;
        t = redu16(c3[r] * qv3); if (ok && nlane == 0) d1[m * 4 + 3] = t;
    }
}

// ---------------------------------------------------------------------------
// GAT layer 2 GEMM: h2[N,32] = hbn[N,64] @ W[64,32]; attention dots (1 head)
// ---------------------------------------------------------------------------
__global__ void gat2_gemm(const float* __restrict__ hbn, const float* __restrict__ W,
                          const float* __restrict__ asrc, const float* __restrict__ adst,
                          float* __restrict__ h2, float* __restrict__ s2,
                          float* __restrict__ d2, int n) {
    int wave = threadIdx.x >> 5;
    int lane = threadIdx.x & 31;
    int tile = blockIdx.x * 8 + wave;
    int ntiles = (n + 15) >> 4;
    if (tile >= ntiles) return;
    int nlane = lane & 15;
    int khalf = lane >> 4;
    int arow = tile * 16 + nlane;
    if (arow >= n) arow = n - 1;

    v8f c0 = {}, c1 = {};
    #pragma unroll
    for (int kk = 0; kk < 16; ++kk) {
        int k = kk * 4 + khalf * 2;
        v2f a = *(const v2f*)(hbn + arow * 64 + k);
        v2f b0, b1;
        b0.x = W[k * 32 +  0 + nlane]; b0.y = W[(k + 1) * 32 +  0 + nlane];
        b1.x = W[k * 32 + 16 + nlane]; b1.y = W[(k + 1) * 32 + 16 + nlane];
        c0 = __builtin_amdgcn_wmma_f32_16x16x4_f32(false, a, false, b0, (short)0, c0, false, false);
        c1 = __builtin_amdgcn_wmma_f32_16x16x4_f32(false, a, false, b1, (short)0, c1, false, false);
    }

    float a0 = asrc[nlane], a1 = asrc[16 + nlane];
    float q0 = adst[nlane], q1 = adst[16 + nlane];

    #pragma unroll
    for (int r = 0; r < 8; ++r) {
        int m = tile * 16 + r + 8 * khalf;
        bool ok = (m < n);
        if (ok) {
            h2[m * 32 +  0 + nlane] = c0[r];
            h2[m * 32 + 16 + nlane] = c1[r];
        }
        float sv = redu16(c0[r] * a0 + c1[r] * a1);
        if (ok && nlane == 0) s2[m] = sv;
        float dv = redu16(c0[r] * q0 + c1[r] * q1);
        if (ok && nlane == 0) d2[m] = dv;
    }
}

// ---------------------------------------------------------------------------
// init: out[n,F] = bias[f]; m = -inf; sum = 0   (F, H powers of two)
// ---------------------------------------------------------------------------
__global__ void init_net(float* __restrict__ out, const float* __restrict__ bias,
                         float* __restrict__ m, float* __restrict__ sum,
                         int n, int F, int H) {
    int idx = blockIdx.x * blockDim.x + threadIdx.x;
    if (idx < n * F) out[idx] = bias[idx & (F - 1)];
    if (idx < n * H) { m[idx] = -INFINITY; sum[idx] = 0.0f; }
}

// ---------------------------------------------------------------------------
// Edge pass 1: segment max of leaky_relu(s[src]+d[dst]) per (dst, head)
// ---------------------------------------------------------------------------
__global__ void edge_max(const int* __restrict__ ei, int E, int n, int Hshift,
                         const float* __restrict__ s, const float* __restrict__ d,
                         float* __restrict__ m) {
    int H = 1 << Hshift;
    int idx = blockIdx.x * blockDim.x + threadIdx.x;
    int total = (E + n) << Hshift;
    if (idx >= total) return;
    int h = idx & (H - 1);
    int e = idx >> Hshift;
    int sn, dn;
    if (e < E) { sn = ei[e]; dn = ei[E + e]; } else { sn = dn = e - E; }
    float v = s[sn * H + h] + d[dn * H + h];
    v = v > 0.0f ? v : NEG_SLOPE * v;
    atomicMaxF(&m[dn * H + h], v);
}

// ---------------------------------------------------------------------------
// Edge pass 2: segment sum of exp(e - m[dst])
// ---------------------------------------------------------------------------
__global__ void edge_sum(const int* __restrict__ ei, int E, int n, int Hshift,
                         const float* __restrict__ s, const float* __restrict__ d,
                         const float* __restrict__ m, float* __restrict__ sum) {
    int H = 1 << Hshift;
    int idx = blockIdx.x * blockDim.x + threadIdx.x;
    int total = (E + n) << Hshift;
    if (idx >= total) return;
    int h = idx & (H - 1);
    int e = idx >> Hshift;
    int sn, dn;
    if (e < E) { sn = ei[e]; dn = ei[E + e]; } else { sn = dn = e - E; }
    float v = s[sn * H + h] + d[dn * H + h];
    v = v > 0.0f ? v : NEG_SLOPE * v;
    float ex = __expf(v - m[dn * H + h]);
    atomicAdd(&sum[dn * H + h], ex);
}

// ---------------------------------------------------------------------------
// Edge pass 3: out[dst,f] += h[src,f] * alpha ; one thread per (edge, feature)
// F = H * HO ; head = f >> HOshift
// ---------------------------------------------------------------------------
__global__ void edge_acc(const int* __restrict__ ei, int E, int n, int Hshift,
                         int Fshift, int HOshift,
                         const float* __restrict__ s, const float* __restrict__ d,
                         const float* __restrict__ m, const float* __restrict__ sum,
                         const float* __restrict__ hsrc, float* __restrict__ out) {
    int H = 1 << Hshift;
    int F = 1 << Fshift;
    int idx = blockIdx.x * blockDim.x + threadIdx.x;
    int total = (E + n) << Fshift;
    if (idx >= total) return;
    int f = idx & (F - 1);
    int e = idx >> Fshift;
    int h = f >> HOshift;
    int sn, dn;
    if (e < E) { sn = ei[e]; dn = ei[E + e]; } else { sn = dn = e - E; }
    float v = s[sn * H + h] + d[dn * H + h];
    v = v > 0.0f ? v : NEG_SLOPE * v;
    float ex = __expf(v - m[dn * H + h]);
    float alpha = ex / (sum[dn * H + h] + 1e-16f);
    atomicAdd(&out[dn * F + f], hsrc[sn * F + f] * alpha);
}

// ---------------------------------------------------------------------------
// BatchNorm (inference) + ReLU, elementwise
// ---------------------------------------------------------------------------
__global__ void bn_relu(const float* __restrict__ in, float* __restrict__ out,
                        const float* __restrict__ g, const float* __restrict__ b,
                        const float* __restrict__ mean, const float* __restrict__ var,
                        int n, int F) {
    int idx = blockIdx.x * blockDim.x + threadIdx.x;
    if (idx >= n * F) return;
    int f = idx & (F - 1);
    float y = (in[idx] - mean[f]) * rsqrtf(var[f] + BN_EPS) * g[f] + b[f];
    out[idx] = fmaxf(y, 0.0f);
}

// ---------------------------------------------------------------------------
// Fused classifier + regressor heads: emb[N,32] -> roles[N,3], energy[N,1]
// ---------------------------------------------------------------------------
__global__ void mlp_heads(const float* __restrict__ emb,
                          const float* __restrict__ cw1, const float* __restrict__ cb1,
                          const float* __restrict__ cw2, const float* __restrict__ cb2,
                          const float* __restrict__ rw1, const float* __restrict__ rb1,
                          const float* __restrict__ rw2, const float* __restrict__ rb2,
                          float* __restrict__ roles, float* __restrict__ energy, int n) {
    __shared__ float s_cw1[1024], s_rw1[1024], s_cw2[96];
    __shared__ float s_cb1[32], s_rb1[32], s_rw2[32], s_cb2[3], s_rb2[1];
    for (int i = threadIdx.x; i < 1024; i += blockDim.x) { s_cw1[i] = cw1[i]; s_rw1[i] = rw1[i]; }
    for (int i = threadIdx.x; i < 96; i += blockDim.x) s_cw2[i] = cw2[i];
    for (int i = threadIdx.x; i < 32; i += blockDim.x) { s_cb1[i] = cb1[i]; s_rb1[i] = rb1[i]; s_rw2[i] = rw2[i]; }
    if (threadIdx.x < 3) s_cb2[threadIdx.x] = cb2[threadIdx.x];
    if (threadIdx.x == 0) s_rb2[0] = rb2[0];
    __syncthreads();

    int node = blockIdx.x * blockDim.x + threadIdx.x;
    if (node >= n) return;
    float e[32];
    #pragma unroll
    for (int i = 0; i < 32; ++i) e[i] = emb[node * 32 + i];

    float r0 = s_cb2[0], r1 = s_cb2[1], r2 = s_cb2[2], en = s_rb2[0];
    #pragma unroll 4
    for (int j = 0; j < 32; ++j) {
        float hc = s_cb1[j], hr = s_rb1[j];
        #pragma unroll
        for (int i = 0; i < 32; ++i) {
            hc += e[i] * s_cw1[i * 32 + j];
            hr += e[i] * s_rw1[i * 32 + j];
        }
        hc = fmaxf(hc, 0.0f);
        hr = fmaxf(hr, 0.0f);
        r0 += hc * s_cw2[j * 3 + 0];
        r1 += hc * s_cw2[j * 3 + 1];
        r2 += hc * s_cw2[j * 3 + 2];
        en += hr * s_rw2[j];
    }
    roles[node * 3 + 0] = r0;
    roles[node * 3 + 1] = r1;
    roles[node * 3 + 2] = r2;
    energy[node] = en;
}

extern "C" void kernel_launch(void* const* d_in, const int* in_sizes, int n_in,
                              void* d_out, int out_size, void* d_ws, size_t ws_size,
                              hipStream_t stream) {
    const float* x        = (const float*)d_in[0];
    const int*   ei       = (const int*)  d_in[1];
    const float* W1       = (const float*)d_in[2];
    const float* att_src1 = (const float*)d_in[3];
    const float* att_dst1 = (const float*)d_in[4];
    const float* b1       = (const float*)d_in[5];
    const float* W2       = (const float*)d_in[6];
    const float* att_src2 = (const float*)d_in[7];
    const float* att_dst2 = (const float*)d_in[8];
    const float* b2       = (const float*)d_in[9];
    const float* bn1_g = (const float*)d_in[10], *bn1_b = (const float*)d_in[11];
    const float* bn1_m = (const float*)d_in[12], *bn1_v = (const float*)d_in[13];
    const float* bn2_g = (const float*)d_in[14], *bn2_b = (const float*)d_in[15];
    const float* bn2_m = (const float*)d_in[16], *bn2_v = (const float*)d_in[17];
    const float* cls_w1 = (const float*)d_in[18], *cls_b1 = (const float*)d_in[19];
    const float* cls_w2 = (const float*)d_in[20], *cls_b2 = (const float*)d_in[21];
    const float* reg_w1 = (const float*)d_in[22], *reg_b1 = (const float*)d_in[23];
    const float* reg_w2 = (const float*)d_in[24], *reg_b2 = (const float*)d_in[25];

    const int n = in_sizes[0] / 32;   // N nodes
    const int E = in_sizes[1] / 2;    // edges before self-loops

    // workspace layout (floats)
    float* ws   = (float*)d_ws;
    float* h1   = ws;                 // [n,64] ; reused as hbn after BN1
    float* s1   = h1   + (size_t)n * 64;  // [n,4]
    float* d1   = s1   + (size_t)n * 4;
    float* m1   = d1   + (size_t)n * 4;
    float* sum1 = m1   + (size_t)n * 4;
    float* out1 = sum1 + (size_t)n * 4;   // [n,64]
    float* h2   = out1 + (size_t)n * 64;  // [n,32]
    float* s2   = h2   + (size_t)n * 32;  // [n]
    float* d2   = s2   + (size_t)n;
    float* m2   = d2   + (size_t)n;
    float* sum2 = m2   + (size_t)n;
    float* out2 = sum2 + (size_t)n;       // [n,32]

    float* emb    = (float*)d_out;            // [n,32]
    float* roles  = emb + (size_t)n * 32;     // [n,3]
    float* energy = roles + (size_t)n * 3;    // [n,1]

    const int TB = 256;
    int ntiles = (n + 15) / 16;
    int gblk   = (ntiles + 7) / 8;

    // ---- layer 1 ----
    gat1_gemm<<<gblk, TB, 0, stream>>>(x, W1, att_src1, att_dst1, h1, s1, d1, n);
    init_net<<<(n * 64 + TB - 1) / TB, TB, 0, stream>>>(out1, b1, m1, sum1, n, 64, 4);
    {
        int totH = (E + n) * 4;
        edge_max<<<(totH + TB - 1) / TB, TB, 0, stream>>>(ei, E, n, 2, s1, d1, m1);
        edge_sum<<<(totH + TB - 1) / TB, TB, 0, stream>>>(ei, E, n, 2, s1, d1, m1, sum1);
        int totF = (E + n) * 64;
        edge_acc<<<(totF + TB - 1) / TB, TB, 0, stream>>>(ei, E, n, 2, 6, 4,
                                                          s1, d1, m1, sum1, h1, out1);
    }
    bn_relu<<<(n * 64 + TB - 1) / TB, TB, 0, stream>>>(out1, h1, bn1_g, bn1_b, bn1_m, bn1_v, n, 64);

    // ---- layer 2 ----
    gat2_gemm<<<gblk, TB, 0, stream>>>(h1, W2, att_src2, att_dst2, h2, s2, d2, n);
    init_net<<<(n * 32 + TB - 1) / TB, TB, 0, stream>>>(out2, b2, m2, sum2, n, 32, 1);
    {
        int totH = (E + n);
        edge_max<<<(totH + TB - 1) / TB, TB, 0, stream>>>(ei, E, n, 0, s2, d2, m2);
        edge_sum<<<(totH + TB - 1) / TB, TB, 0, stream>>>(ei, E, n, 0, s2, d2, m2, sum2);
        int totF = (E + n) * 32;
        edge_acc<<<(totF + TB - 1) / TB, TB, 0, stream>>>(ei, E, n, 0, 5, 5,
                                                          s2, d2, m2, sum2, h2, out2);
    }
    bn_relu<<<(n * 32 + TB - 1) / TB, TB, 0, stream>>>(out2, emb, bn2_g, bn2_b, bn2_m, bn2_v, n, 32);

    // ---- heads ----
    mlp_heads<<<(n + TB - 1) / TB, TB, 0, stream>>>(emb, cls_w1, cls_b1, cls_w2, cls_b2,
                                                    reg_w1, reg_b1, reg_w2, reg_b2,
                                                    roles, energy, n);
}